// Eq2to2_69372311765323
// MI455X (gfx1250) — compile-verified
//
#include <hip/hip_runtime.h>

typedef __attribute__((ext_vector_type(2))) float v2f;
typedef __attribute__((ext_vector_type(8))) float v8f;

// ---------------- workspace layout (float offsets) ----------------
// Agg  [512][384] : cols 0:128 = sum_cols(j), 128:256 = sum_rows(i), 256:384 = diag(i)
// H    [512][384] : cols 0:128 = Fi(i-terms), 128:256 = Fj(j-terms), 256:384 = G(diag terms)
// Wbig : 96*2*384 pairs (WMMA-B interleaved), W10/W11 : 32*2*128 pairs each
enum : size_t {
  AGG_OFF  = 0,
  H_OFF    = 196608,
  WBIG_OFF = 393216,
  W10_OFF  = 540672,
  W11_OFF  = 557056,
  SD_OFF   = 573440,   // sum_diag [4][128]
  SA_OFF   = 573952,   // sum_all  [4][128]
  SFI_OFF  = 574464,   // per-(n,s) scalar added everywhere (incl. bias)
  SG_OFF   = 574976,   // per-(n,s) scalar added on diagonal (incl. diag_bias)
};

__device__ __forceinline__ v8f wmma_f32(v2f a, v2f b, v8f c) {
  // V_WMMA_F32_16X16X4_F32 : D = A(16x4 f32) * B(4x16 f32) + C(16x16 f32)
  return __builtin_amdgcn_wmma_f32_16x16x4_f32(
      /*neg_a=*/false, a, /*neg_b=*/false, b,
      /*c_mod=*/(short)0, c, /*reuse_a=*/false, /*reuse_b=*/false);
}

// ---------- K1a: sum over j (rows) + diagonal, per (n,i), thread = d ----------
__global__ __launch_bounds__(128) void k_rows_diag(const float* __restrict__ x,
                                                   float* __restrict__ ws) {
  const int p = blockIdx.x;        // n*128 + i
  const int i = p & 127;
  const int d = threadIdx.x;
  const float* xb = x + ((size_t)p * 128) * 128 + d;   // x[n,i,j,d], j varies
  float acc = 0.f, dg = 0.f;
  for (int j = 0; j < 128; ++j) {
    float v = xb[(size_t)j * 128];
    acc += v;
    if (j == i) dg = v;
  }
  float* agg = ws + AGG_OFF + (size_t)p * 384;
  agg[128 + d] = acc;   // sum_rows[n,i,d]
  agg[256 + d] = dg;    // diag[n,i,d]
}

// ---------- K1b: sum over i (cols), per (n,j), thread = d ----------
__global__ __launch_bounds__(128) void k_cols(const float* __restrict__ x,
                                              float* __restrict__ ws) {
  const int p = blockIdx.x;        // n*128 + j
  const int n = p >> 7, j = p & 127;
  const int d = threadIdx.x;
  const float* xb = x + (((size_t)n * 128) * 128 + j) * 128 + d;
  float acc = 0.f;
  for (int i = 0; i < 128; ++i) acc += xb[(size_t)i * 128 * 128];
  ws[AGG_OFF + (size_t)p * 384 + d] = acc;   // sum_cols[n,j,d]
}

// ---------- K1c: sum_diag / sum_all per (n,d) ----------
__global__ __launch_bounds__(128) void k_nd_sums(float* __restrict__ ws) {
  const int n = blockIdx.x, d = threadIdx.x;
  float sd = 0.f, sa = 0.f;
  for (int i = 0; i < 128; ++i) {
    const float* agg = ws + AGG_OFF + (size_t)(n * 128 + i) * 384;
    sa += agg[128 + d];   // sum over i of row-sums = total sum
    sd += agg[256 + d];   // sum of diagonal
  }
  ws[SD_OFF + n * 128 + d] = sd;
  ws[SA_OFF + n * 128 + d] = sa;
}

// ---------- K_scal: per-(n,s) scalar terms (ops 2,5,14,15) + biases ----------
__global__ __launch_bounds__(128) void k_scal(const float* __restrict__ coefs,
                                              const float* __restrict__ bias,
                                              const float* __restrict__ dbias,
                                              float* __restrict__ ws) {
  const int n = blockIdx.x, s = threadIdx.x;
  float aFi = bias[s], aG = dbias[s];
  for (int d = 0; d < 128; ++d) {
    const float* c = coefs + ((size_t)d * 128 + s) * 15;
    const float sd = ws[SD_OFF + n * 128 + d];
    const float sa = ws[SA_OFF + n * 128 + d];
    aFi += c[13] * sd + c[14] * sa;   // ops 14,15: everywhere
    aG  += c[1]  * sd + c[4]  * sa;   // ops 2,5  : diagonal only
  }
  ws[SFI_OFF + n * 128 + s] = aFi;
  ws[SG_OFF  + n * 128 + s] = aG;
}

// ---------- coef packing into WMMA-B interleaved layout ----------
__device__ __forceinline__ float wbig_val(const float* __restrict__ coefs, int k, int c) {
  const int kb = k >> 7, kd = k & 127;   // feature block: 0=sum_cols,1=sum_rows,2=diag
  const int cb = c >> 7, cs = c & 127;   // output block : 0=Fi,1=Fj,2=G
  const int btab[9] = {7, 5, 3,    // sc -> {Fi:op8, Fj:op6, G:op4}
                       8, 6, 2,    // sr -> {Fi:op9, Fj:op7, G:op3}
                       11, 12, 0}; // dg -> {Fi:op12,Fj:op13,G:op1}
  return coefs[((size_t)kd * 128 + cs) * 15 + btab[kb * 3 + cb]];
}

__global__ void k_pack(const float* __restrict__ coefs, float* __restrict__ ws) {
  const int t = blockIdx.x * blockDim.x + threadIdx.x;
  const int NBIG = 96 * 2 * 384;   // 73728 pairs
  const int NS   = 32 * 2 * 128;   // 8192 pairs
  if (t < NBIG) {
    const int kk = t / 768, r2 = t % 768, h = r2 / 384, c = r2 % 384;
    const int k0 = kk * 4 + h * 2;
    v2f w; w.x = wbig_val(coefs, k0, c); w.y = wbig_val(coefs, k0 + 1, c);
    *(v2f*)(ws + WBIG_OFF + (size_t)t * 2) = w;
  } else if (t < NBIG + NS) {
    const int u = t - NBIG;
    const int kk = u / 256, r2 = u % 256, h = r2 / 128, s = r2 % 128;
    const int d0 = kk * 4 + h * 2;
    v2f w;
    w.x = coefs[((size_t)d0 * 128 + s) * 15 + 9];       // op10: x[i,j]
    w.y = coefs[((size_t)(d0 + 1) * 128 + s) * 15 + 9];
    *(v2f*)(ws + W10_OFF + (size_t)u * 2) = w;
  } else if (t < NBIG + 2 * NS) {
    const int u = t - NBIG - NS;
    const int kk = u / 256, r2 = u % 256, h = r2 / 128, s = r2 % 128;
    const int d0 = kk * 4 + h * 2;
    v2f w;
    w.x = coefs[((size_t)d0 * 128 + s) * 15 + 10];      // op11: x[j,i]
    w.y = coefs[((size_t)(d0 + 1) * 128 + s) * 15 + 10];
    *(v2f*)(ws + W11_OFF + (size_t)u * 2) = w;
  }
}

// ---------- K2: H[512,384] = Agg[512,384] x Wbig[384,384] via WMMA ----------
__global__ __launch_bounds__(256) void k_gemm_agg(float* __restrict__ ws) {
  const int lane = threadIdx.x & 31, wv = threadIdx.x >> 5;
  const int tile = blockIdx.x * 8 + wv;        // [0,768)
  const int rt = tile / 24, ct = tile % 24;
  const int p0 = rt * 16, c0 = ct * 16;
  const int r = lane & 15, h = lane >> 4;
  const float* A  = ws + AGG_OFF;
  const float* Bm = ws + WBIG_OFF;
  v8f acc = {0.f, 0.f, 0.f, 0.f, 0.f, 0.f, 0.f, 0.f};
  for (int k = 0; k < 96; ++k) {
    v2f a = *(const v2f*)(A + (size_t)(p0 + r) * 384 + k * 4 + 2 * h);
    v2f b = *(const v2f*)(Bm + ((size_t)(k * 2 + h) * 384 + c0 + r) * 2);
    acc = wmma_f32(a, b, acc);
  }
  float* H = ws + H_OFF;
#pragma unroll
  for (int v = 0; v < 8; ++v)
    H[(size_t)(p0 + v + 8 * h) * 384 + c0 + r] = acc[v];
}

// ---------- K3: main fused kernel ----------
// wave tile = (n, i, j0..j0+15) x all 128 s channels (8 accumulators).
// out = leakyrelu( x[i,j,:]W10 + x[j,i,:]W11 + Fi[n,i,s]+sFi[n,s] + Fj[n,j,s]
//                  + (i==j)(G[n,i,s]+sG[n,s]) )
__global__ __launch_bounds__(256) void k_main(const float* __restrict__ x,
                                              float* __restrict__ out,
                                              const float* __restrict__ ws) {
  const int lane = threadIdx.x & 31, wv = threadIdx.x >> 5;
  const int tile = blockIdx.x * 8 + wv;        // [0,4096)
  const int n   = tile >> 10;
  const int rem = tile & 1023;
  const int i   = rem >> 3;
  const int j0  = (rem & 7) << 4;
  const int r = lane & 15, h = lane >> 4;
  const size_t ni = (size_t)(n * 128 + i);
  const float* W10 = ws + W10_OFF;
  const float* W11 = ws + W11_OFF;
  const float* xd = x + (ni * 128 + (size_t)(j0 + r)) * 128;                     // A row (direct)
  const float* xt = x + (((size_t)(n * 128 + j0 + r)) * 128 + (size_t)i) * 128;  // A row (transposed)

  v8f acc[8];
#pragma unroll
  for (int sc = 0; sc < 8; ++sc)
    acc[sc] = (v8f){0.f, 0.f, 0.f, 0.f, 0.f, 0.f, 0.f, 0.f};

  for (int k = 0; k < 32; ++k) {
    const int d0 = k * 4 + 2 * h;
    v2f ad = *(const v2f*)(xd + d0);
    v2f at = *(const v2f*)(xt + d0);
    const int bidx = (k * 2 + h) * 128;
#pragma unroll
    for (int sc = 0; sc < 8; ++sc) {
      const int s = sc * 16 + r;
      v2f b0 = *(const v2f*)(W10 + (size_t)(bidx + s) * 2);
      v2f b1 = *(const v2f*)(W11 + (size_t)(bidx + s) * 2);
      acc[sc] = wmma_f32(ad, b0, acc[sc]);
      acc[sc] = wmma_f32(at, b1, acc[sc]);
    }
  }

  const float* H   = ws + H_OFF;
  const float* sFi = ws + SFI_OFF + n * 128;
  const float* sG  = ws + SG_OFF  + n * 128;
#pragma unroll
  for (int sc = 0; sc < 8; ++sc) {
    const int s = sc * 16 + r;
    const float fi = H[ni * 384 + s] + sFi[s];
    const float g  = H[ni * 384 + 256 + s] + sG[s];
#pragma unroll
    for (int v = 0; v < 8; ++v) {
      const int j = j0 + v + 8 * h;
      float val = acc[sc][v] + fi + H[(size_t)(n * 128 + j) * 384 + 128 + s];
      if (j == i) val += g;
      val = (val > 0.f) ? val : val * 0.01f;
      out[(ni * 128 + (size_t)j) * 128 + s] = val;
    }
  }
}

extern "C" void kernel_launch(void* const* d_in, const int* in_sizes, int n_in,
                              void* d_out, int out_size, void* d_ws, size_t ws_size,
                              hipStream_t stream) {
  const float* x     = (const float*)d_in[0];   // [4,128,128,128]
  const float* coefs = (const float*)d_in[1];   // [128,128,15]
  const float* bias  = (const float*)d_in[2];   // [128]
  const float* dbias = (const float*)d_in[3];   // [128]
  float* out = (float*)d_out;
  float* ws  = (float*)d_ws;                    // needs ~2.31 MB

  k_rows_diag<<<512, 128, 0, stream>>>(x, ws);
  k_cols     <<<512, 128, 0, stream>>>(x, ws);
  k_pack     <<<352, 256, 0, stream>>>(coefs, ws);   // 352*256 == 90112 pack slots
  k_nd_sums  <<<4,   128, 0, stream>>>(ws);
  k_scal     <<<4,   128, 0, stream>>>(coefs, bias, dbias, ws);
  k_gemm_agg <<<96,  256, 0, stream>>>(ws);
  k_main     <<<512, 256, 0, stream>>>(x, out, ws);
}